// PackedStdScaler_12489764897046
// MI455X (gfx1250) — compile-verified
//
#include <hip/hip_runtime.h>
#include <math.h>

// ---------------------------------------------------------------------------
// PackedStdScaler on MI455X (gfx1250).
// Pairwise group-mean/variance mapped onto V_WMMA_F32_16X16X4_F32:
//   per 16-row s1 tile, accumulate C[16x16] += A(16x4 of w) * B(4x16 where
//   col0 = t, col1 = 1, col2 = t^2) over all s2 in steps of 4.
// B operands are built with packed FMAs from loop-invariant per-lane column
// selectors (bmat = t^2*c2 + t*c0 + c1), so the inner loop is
// ds_load + 2x(v_cmp,v_cndmask) + ~3 pk-VALU + 1 WMMA per K=4 chunk.
// LDS stages the batch row (keys + t) once per block; epilogue extracts
// columns 0..2 of the f32 C layout and finishes loc/scale.
// ---------------------------------------------------------------------------

typedef float v2f __attribute__((ext_vector_type(2)));
typedef float v8f __attribute__((ext_vector_type(8)));

#define BB 4
#define SS 2048
#define TILES_PER_BLOCK 4
#define THREADS (TILES_PER_BLOCK * 32)

// Pre-pass: pack (sample_id, variate_id, depth_val) into a single int key.
// key_raw   : raw key (used for the s1 side of the comparison)
// key_mask  : key if mask[s2] else -1 (s2 side; -1 never matches a raw key)
__global__ void pack_keys_kernel(const unsigned char* __restrict__ mask,
                                 const int* __restrict__ sid,
                                 const int* __restrict__ vid,
                                 const float* __restrict__ dv,
                                 int* __restrict__ key_mask,
                                 int* __restrict__ key_raw,
                                 int n) {
  int i = blockIdx.x * blockDim.x + threadIdx.x;
  if (i < n) {
    int k = sid[i] * 32 + vid[i] * 4 + (int)dv[i];  // sid<64, vid<8, dv in {0..3}
    key_raw[i]  = k;
    key_mask[i] = mask[i] ? k : -1;
  }
}

__global__ void __launch_bounds__(THREADS)
packed_std_scaler_kernel(const float* __restrict__ target,
                         const int* __restrict__ sid,
                         const int* __restrict__ key_mask,
                         const int* __restrict__ key_raw,
                         float* __restrict__ loc_out,
                         float* __restrict__ scale_out) {
  __shared__ int   s_key[SS];
  __shared__ float s_t[SS];
  __shared__ float s_acc[THREADS * 8];

  const int tid  = threadIdx.x;
  const int wave = tid >> 5;
  const int lane = tid & 31;

  const int blocksPerBatch = (SS / 16) / TILES_PER_BLOCK;  // 32
  const int b     = blockIdx.x / blocksPerBatch;
  const int tile0 = (blockIdx.x % blocksPerBatch) * TILES_PER_BLOCK;

  // Stage this batch row once: 2048 keys + 2048 t values -> 16 KB LDS.
  for (int i = tid; i < SS; i += THREADS) {
    s_key[i] = key_mask[b * SS + i];
    s_t[i]   = target[b * SS + i];
  }
  __syncthreads();

  const int myTile = tile0 + wave;
  const int row    = lane & 15;                 // M index for A, N index for B
  const int s1     = myTile * 16 + row;
  const int myKey  = key_raw[b * SS + s1];      // raw key of this s1 row

  // A 16x4 f32 layout: lanes 0-15 hold K=0 (v0) / K=1 (v1) for M=lane;
  // lanes 16-31 hold K=2 / K=3. B 4x16 mirrors with N = lane&15.
  const int koff = (lane < 16) ? 0 : 2;
  const int col  = lane & 15;

  // Loop-invariant column selectors: B[k, col] = t^2*c2 + t*c0 + c1.
  const float c0 = (col == 0) ? 1.0f : 0.0f;   // column 0: t
  const float c1 = (col == 1) ? 1.0f : 0.0f;   // column 1: 1
  const float c2 = (col == 2) ? 1.0f : 0.0f;   // column 2: t^2
  const v2f c0v = {c0, c0};
  const v2f c1v = {c1, c1};
  const v2f c2v = {c2, c2};

  v8f acc = {0.f, 0.f, 0.f, 0.f, 0.f, 0.f, 0.f, 0.f};

#pragma unroll 8
  for (int it = 0; it < SS / 4; ++it) {
    const int s2 = it * 4 + koff;
    const int k0 = s_key[s2];
    const int k1 = s_key[s2 + 1];
    v2f t;
    t.x = s_t[s2];
    t.y = s_t[s2 + 1];

    v2f a;
    a.x = (k0 == myKey) ? 1.0f : 0.0f;
    a.y = (k1 == myKey) ? 1.0f : 0.0f;

    // pk_mul + 2x pk_fma; branch-free so EXEC stays all-ones for the WMMA.
    const v2f bmat = t * t * c2v + t * c0v + c1v;

    acc = __builtin_amdgcn_wmma_f32_16x16x4_f32(
        /*neg_a=*/false, a, /*neg_b=*/false, bmat,
        /*c_mod=*/(short)0, acc, /*reuse_a=*/false, /*reuse_b=*/false);
  }

  // Spill accumulators so lanes 0-15 can gather columns 0..2.
#pragma unroll
  for (int v = 0; v < 8; ++v) s_acc[tid * 8 + v] = acc[v];
  __syncthreads();

  if (lane < 16) {
    const int r    = lane;              // output row M
    const int base = wave * 32 * 8;
    const int hi   = (r >= 8) ? 16 : 0; // M>=8 lives in lanes 16-31
    const int vg   = r & 7;             // VGPR index = M % 8
    // C element (M, N): value at lane hi+N, vgpr M%8.
    const float sum_wt = s_acc[base + (hi + 0) * 8 + vg];  // N=0: sum w*t
    const float cnt    = s_acc[base + (hi + 1) * 8 + vg];  // N=1: sum w
    const float sumsq  = s_acc[base + (hi + 2) * 8 + vg];  // N=2: sum w*t^2

    const float loc = sum_wt / ((cnt == 0.0f) ? 1.0f : cnt);      // safe_div
    const float num = sumsq - 2.0f * loc * sum_wt + cnt * loc * loc;
    const float den = cnt - 1.0f;                                  // CORRECTION
    float var = num / ((den == 0.0f) ? 1.0f : den);                // safe_div
    var = fmaxf(var, 0.0f);
    float scl = sqrtf(var + 1e-5f);                                // MINIMUM_SCALE

    const int idx = b * SS + myTile * 16 + r;
    float loc_o = loc, scl_o = scl;
    if (sid[idx] == 0) { loc_o = 0.0f; scl_o = 1.0f; }             // g0 override
    loc_out[idx]   = loc_o;
    scale_out[idx] = scl_o;
  }
}

extern "C" void kernel_launch(void* const* d_in, const int* in_sizes, int n_in,
                              void* d_out, int out_size, void* d_ws, size_t ws_size,
                              hipStream_t stream) {
  // setup_inputs order: target, mask, sample_id, variate_id, depth_val
  const float*         target = (const float*)d_in[0];
  const unsigned char* mask   = (const unsigned char*)d_in[1];  // bool -> 1 byte
  const int*           sid    = (const int*)d_in[2];
  const int*           vid    = (const int*)d_in[3];
  const float*         dv     = (const float*)d_in[4];

  float* out   = (float*)d_out;          // [loc (B*S) | scale (B*S)]
  int*   key_m = (int*)d_ws;             // masked keys (s2 side)
  int*   key_r = key_m + BB * SS;        // raw keys (s1 side)

  const int n = BB * SS;
  pack_keys_kernel<<<(n + 255) / 256, 256, 0, stream>>>(mask, sid, vid, dv,
                                                        key_m, key_r, n);

  const int grid = BB * (SS / 16) / TILES_PER_BLOCK;  // 128 blocks, 4 waves each
  packed_std_scaler_kernel<<<grid, THREADS, 0, stream>>>(
      target, sid, key_m, key_r, out, out + n);
}